// MPN_simplenet_70574902608021
// MI455X (gfx1250) — compile-verified
//
#include <hip/hip_runtime.h>
#include <math.h>

#define NNODES 50000
#define NEDGES 800000

typedef __attribute__((ext_vector_type(2))) float v2f;
typedef __attribute__((ext_vector_type(8))) float v8f;

// ---------------- utility kernels ----------------

__global__ void zero_f32(float* __restrict__ p, int n) {
  int i = blockIdx.x * blockDim.x + threadIdx.x;
  if (i < n) p[i] = 0.0f;
}

// int64 edge index -> int32 row/col, and accumulate target degree
__global__ void edge_prep(const long long* __restrict__ ei, int* __restrict__ row32,
                          int* __restrict__ col32, float* __restrict__ deg, int E) {
  int e = blockIdx.x * blockDim.x + threadIdx.x;
  if (e >= E) return;
  int r = (int)ei[e];          // source j
  int c = (int)ei[E + e];      // target i
  row32[e] = r;
  col32[e] = c;
  unsafeAtomicAdd(&deg[c], 1.0f);  // global_atomic_add_f32, L2-resident
}

__global__ void deg_to_dinv(float* __restrict__ p, int n) {
  int i = blockIdx.x * blockDim.x + threadIdx.x;
  if (i >= n) return;
  float d = p[i];
  p[i] = (d > 0.0f) ? (1.0f / sqrtf(d)) : 0.0f;
}

__global__ void edge_norm(const int* __restrict__ row, const int* __restrict__ col,
                          const float* __restrict__ dinv, float* __restrict__ nrm, int E) {
  int e = blockIdx.x * blockDim.x + threadIdx.x;
  if (e >= E) return;
  nrm[e] = dinv[row[e]] * dinv[col[e]];
}

// out[col[e]] += norm[e] * h[row[e]]  — one thread per (edge, 4-channel chunk)
__global__ void prop_scatter(const float* __restrict__ h, const int* __restrict__ row,
                             const int* __restrict__ col, const float* __restrict__ nrm,
                             float* out, int tpe_shift, int E) {
  int tid = blockIdx.x * blockDim.x + threadIdx.x;
  int e = tid >> tpe_shift;
  if (e >= E) return;
  int c = (tid & ((1 << tpe_shift) - 1)) << 2;   // channel offset (x4)
  int d = 4 << tpe_shift;                        // feature width (16 or 64)
  int r = row[e];
  int t = col[e];
  float w = nrm[e];
  float4 v = *(const float4*)(h + (size_t)r * d + c);
  float* o = out + (size_t)t * d + c;
  unsafeAtomicAdd(o + 0, w * v.x);
  unsafeAtomicAdd(o + 1, w * v.y);
  unsafeAtomicAdd(o + 2, w * v.z);
  unsafeAtomicAdd(o + 3, w * v.w);
}

// ---------------- weight repack into B-fragment lane order ----------------
// Wp index: (((hop*(K/4) + kq)*NT + t)*32 + lane)*2 holds the v2f each lane
// feeds to V_WMMA_F32_16X16X4_F32 (rows K=kq*4+kg, kg+1; col n=t*16+l16).
// Columns >= N are zero-padded so the GEMM inner loop needs no guards.

__global__ void w_repack(const float* __restrict__ W, float* __restrict__ Wp,
                         int K, int N, int NT, int total) {
  int id = blockIdx.x * blockDim.x + threadIdx.x;
  if (id >= total) return;
  int lane = id & 31;
  int rest = id >> 5;
  int t = rest % NT;  rest /= NT;
  int kq = rest % (K >> 2);  rest /= (K >> 2);
  int hop = rest;
  int half = lane >> 4;
  int l16 = lane & 15;
  int kk = (kq << 2) + (half << 1);
  int n = (t << 4) + l16;
  float v0 = 0.0f, v1 = 0.0f;
  if (n < N) {
    const float* w = W + (size_t)hop * K * N + (size_t)kk * N + n;
    v0 = w[0];
    v1 = w[N];
  }
  Wp[(size_t)id * 2]     = v0;
  Wp[(size_t)id * 2 + 1] = v1;
}

// ---------------- fused TAGConv layer ----------------
// out = [relu](h0@W0 + h1@W1 + h2@W2 + h3@W3 + b)
// One wave per 16-row M tile; compile-time NT/N/K => branch-free inner loop:
// 1x global_load_b64 (A) + NT x global_load_b64 (B) + NT x wmma per k-step.

template <int NT, int N, int K>
__global__ void tag_gemm(const float* h0, const float* h1,
                         const float* h2, const float* h3,
                         const float* __restrict__ Wp, const float* __restrict__ bias,
                         float* out, int do_relu, int ntiles) {
  int wave = (int)((blockIdx.x * blockDim.x + threadIdx.x) >> 5);
  if (wave >= ntiles) return;                 // wave-uniform: EXEC stays all-ones
  int lane = threadIdx.x & 31;
  int half = lane >> 4;                       // 0 or 1
  int l16  = lane & 15;
  int m0   = wave << 4;
  int kg   = half << 1;                       // K sub-offset 0 or 2

  v8f acc[NT];
#pragma unroll
  for (int t = 0; t < NT; ++t) acc[t] = (v8f)0.0f;

  const float* Hs[4] = {h0, h1, h2, h3};
  for (int hop = 0; hop < 4; ++hop) {
    const float* H  = Hs[hop] + (size_t)(m0 + l16) * K;
    const float* Wb = Wp + (size_t)hop * (K / 4) * NT * 64 + lane * 2;
#pragma unroll
    for (int kq = 0; kq < K / 4; ++kq) {
      v2f a = *(const v2f*)(H + (kq << 2) + kg);
#pragma unroll
      for (int t = 0; t < NT; ++t) {
        v2f b = *(const v2f*)(Wb + (size_t)(kq * NT + t) * 64);
        acc[t] = __builtin_amdgcn_wmma_f32_16x16x4_f32(
            false, a, false, b, (short)0, acc[t], false, false);
      }
    }
  }

  // Epilogue: C/D layout — VGPR g holds row m0+g (lanes 0-15) / m0+g+8 (lanes 16-31)
#pragma unroll
  for (int t = 0; t < NT; ++t) {
    int n = (t << 4) + l16;
    if (n < N) {
      float bv = bias[n];
#pragma unroll
      for (int g = 0; g < 8; ++g) {
        int r = m0 + g + (half << 3);
        float v = acc[t][g] + bv;
        if (do_relu) v = fmaxf(v, 0.0f);
        out[(size_t)r * N + n] = v;
      }
    }
  }
}

// ---------------- launch ----------------

extern "C" void kernel_launch(void* const* d_in, const int* in_sizes, int n_in,
                              void* d_out, int out_size, void* d_ws, size_t ws_size,
                              hipStream_t stream) {
  const float* x = (const float*)d_in[0];
  const long long* ei = (const long long*)d_in[1];
  const float* Wl[5];
  const float* bl[5];
  for (int i = 0; i < 5; ++i) {
    Wl[i] = (const float*)d_in[2 + 2 * i];
    bl[i] = (const float*)d_in[3 + 2 * i];
  }

  const int E = NEDGES, NN = NNODES;

  // workspace layout (all 16B aligned by construction)
  int*   row32 = (int*)d_ws;
  int*   col32 = row32 + E;
  float* nrm   = (float*)(col32 + E);
  float* deg   = nrm + E;                      // becomes dinv in place
  float* h1    = deg + NN;
  float* h2    = h1 + (size_t)NN * 64;
  float* h3    = h2 + (size_t)NN * 64;
  float* hio   = h3 + (size_t)NN * 64;         // layer input/output (in-place safe per-tile)
  float* wpack = hio + (size_t)NN * 64;        // <= 64KB fragment-packed weights

  const int TB = 256;

  // --- preprocessing: norm = dinv[row]*dinv[col]
  zero_f32<<<(NN + TB - 1) / TB, TB, 0, stream>>>(deg, NN);
  edge_prep<<<(E + TB - 1) / TB, TB, 0, stream>>>(ei, row32, col32, deg, E);
  deg_to_dinv<<<(NN + TB - 1) / TB, TB, 0, stream>>>(deg, NN);
  edge_norm<<<(E + TB - 1) / TB, TB, 0, stream>>>(row32, col32, deg, nrm, E);

  const int din[5]  = {16, 64, 64, 64, 64};
  const int dout[5] = {64, 64, 64, 64, 4};
  const float* hin = x;
  float* hops[3] = {h1, h2, h3};

  const int ntiles = NN / 16;                        // 3125 (M divides 16 exactly)
  const int gblocks = (ntiles * 32 + TB - 1) / TB;   // one wave per tile

  for (int l = 0; l < 5; ++l) {
    int K = din[l], Nn = dout[l];
    int NT = (Nn > 16) ? 4 : 1;
    int shift = (K == 16) ? 2 : 4;                   // threads per edge = K/4

    // 3 propagation hops: h1 = A h_in, h2 = A h1, h3 = A h2
    const float* src = hin;
    for (int p = 0; p < 3; ++p) {
      int nelem = NN * K;
      zero_f32<<<(nelem + TB - 1) / TB, TB, 0, stream>>>(hops[p], nelem);
      int threads = E << shift;
      prop_scatter<<<(threads + TB - 1) / TB, TB, 0, stream>>>(
          src, row32, col32, nrm, hops[p], shift, E);
      src = hops[p];
    }

    // repack this layer's weights into B-fragment order (zero-padded)
    int total = 4 * (K / 4) * NT * 32;
    w_repack<<<(total + TB - 1) / TB, TB, 0, stream>>>(Wl[l], wpack, K, Nn, NT, total);

    float* outp = (l == 4) ? (float*)d_out : hio;
    int relu = (l < 4) ? 1 : 0;
    if (l == 0)
      tag_gemm<4, 64, 16><<<gblocks, TB, 0, stream>>>(hin, h1, h2, h3, wpack, bl[l],
                                                      outp, relu, ntiles);
    else if (l < 4)
      tag_gemm<4, 64, 64><<<gblocks, TB, 0, stream>>>(hin, h1, h2, h3, wpack, bl[l],
                                                      outp, relu, ntiles);
    else
      tag_gemm<1, 4, 64><<<gblocks, TB, 0, stream>>>(hin, h1, h2, h3, wpack, bl[l],
                                                     outp, relu, ntiles);
    hin = hio;
  }
}